// DetectionLoss_50319836840348
// MI455X (gfx1250) — compile-verified
//
#include <hip/hip_runtime.h>
#include <cstdint>

// ---------------------------------------------------------------------------
// Detection loss (YOLO-style) for MI455X / gfx1250, wave32.
//   kernel 1: per-anchor IoU-match + obj BCE + GIoU + focal class loss,
//             targets staged to LDS via CDNA5 async global->LDS DMA,
//             packed float4 corner reads (1x ds_load_b128 + 1x ds_load_b32
//             per target in the hot loop), deterministic per-block partials.
//   kernel 2: deterministic tree reduction + final scalar math.
// ---------------------------------------------------------------------------

namespace {
constexpr int   CNUM      = 80;     // NUM_CLASSES
constexpr int   MT        = 100;    // targets per image
constexpr int   BLK       = 256;    // 8 waves (wave32)
constexpr float LAM_NOOBJ = 0.5f;
constexpr float LAM_BBOX  = 5.0f;
constexpr float IOU_POS   = 0.5f;
constexpr float ALPHA     = 0.25f;
constexpr float EPSF      = 1e-6f;
}

__device__ __forceinline__ float bce_logits(float x, float t) {
    // max(x,0) - x*t + log1p(exp(-|x|))
    return fmaxf(x, 0.0f) - x * t + log1pf(__expf(-fabsf(x)));
}

__global__ void __launch_bounds__(BLK)
det_loss_main(const float* __restrict__ pred_raw,
              const float* __restrict__ pred_boxes,
              const float* __restrict__ tgt_boxes,
              const int*   __restrict__ tgt_labels,
              float*       __restrict__ partials,
              int N, int blocksPerImg)
{
    __shared__ __align__(16) float  tb[MT * 4];   // raw cx,cy,w,h per target
    __shared__ __align__(16) int    tl[MT];       // labels
    __shared__ __align__(16) float4 tcor[MT];     // packed corners x1,y1,x2,y2
    __shared__ float  tarea[MT];
    __shared__ float  red[(BLK / 32) * 4];

    const int tid = threadIdx.x;
    const int b   = blockIdx.x / blocksPerImg;
    const int nb  = blockIdx.x % blocksPerImg;
    const int n   = nb * BLK + tid;

    // ---- stage this image's targets into LDS via async DMA (ASYNCcnt path)
    {
        const float* gboxes = tgt_boxes  + (size_t)b * MT * 4;  // 1600 B, 16B aligned
        const int*   glab   = tgt_labels + (size_t)b * MT;      //  400 B, 16B aligned
        if (tid < MT) {
            uint32_t lds_off = (uint32_t)(uintptr_t)(&tb[tid * 4]);
            uint32_t goff    = (uint32_t)tid * 16u;
            asm volatile("global_load_async_to_lds_b128 %0, %1, %2"
                         :: "v"(lds_off), "v"(goff), "s"(gboxes) : "memory");
        }
        if (tid < MT / 4) {   // 25 lanes x 16 B = 400 B of labels
            uint32_t lds_off = (uint32_t)(uintptr_t)(&tl[tid * 4]);
            uint32_t goff    = (uint32_t)tid * 16u;
            asm volatile("global_load_async_to_lds_b128 %0, %1, %2"
                         :: "v"(lds_off), "v"(goff), "s"(glab) : "memory");
        }
        asm volatile("s_wait_asynccnt 0" ::: "memory");
    }
    __syncthreads();

    // precompute packed target corners + areas (hot loop reads are broadcasts)
    if (tid < MT) {
        float cx = tb[tid * 4 + 0], cy = tb[tid * 4 + 1];
        float w  = tb[tid * 4 + 2], h  = tb[tid * 4 + 3];
        tcor[tid]  = make_float4(cx - 0.5f * w, cy - 0.5f * h,
                                 cx + 0.5f * w, cy + 0.5f * h);
        tarea[tid] = w * h;
    }
    __syncthreads();

    float obj_sum = 0.0f, bbox_sum = 0.0f, cls_sum = 0.0f, npos = 0.0f;

    if (n < N) {
        const float4 pb = ((const float4*)pred_boxes)[(size_t)b * N + n];
        const float pw = pb.z, ph = pb.w;
        const float p1x = pb.x - 0.5f * pw, p1y = pb.y - 0.5f * ph;
        const float p2x = pb.x + 0.5f * pw, p2y = pb.y + 0.5f * ph;
        const float pa  = pw * ph;

        // ---- max/argmax IoU over targets; division-free cross-mult compare
        float best_in, best_un;
        {
            const float4 t0 = tcor[0];
            float iw = fmaxf(fminf(p2x, t0.z) - fmaxf(p1x, t0.x), 0.0f);
            float ih = fmaxf(fminf(p2y, t0.w) - fmaxf(p1y, t0.y), 0.0f);
            best_in = iw * ih;
            best_un = pa + tarea[0] - best_in;
        }
        int best_m = 0;
        #pragma unroll 4
        for (int m = 1; m < MT; ++m) {
            const float4 t = tcor[m];                 // one ds_load_b128 (broadcast)
            float iw = fmaxf(fminf(p2x, t.z) - fmaxf(p1x, t.x), 0.0f);
            float ih = fmaxf(fminf(p2y, t.w) - fmaxf(p1y, t.y), 0.0f);
            float in = iw * ih;
            float un = pa + tarea[m] - in;            // one ds_load_b32 (broadcast)
            // in/(un+eps) > best_in/(best_un+eps); denominators > 0
            if (in * (best_un + EPSF) > best_in * (un + EPSF)) {
                best_in = in; best_un = un; best_m = m;
            }
        }
        const float max_iou = best_in / (best_un + EPSF);
        const bool  pos     = (max_iou >= IOU_POS);
        const float posf    = pos ? 1.0f : 0.0f;

        // ---- objectness BCE (read only channel 0 of pred_raw)
        const size_t praw = ((size_t)b * N + n) * (size_t)(5 + CNUM);
        const float  ox   = pred_raw[praw];
        obj_sum = bce_logits(ox, posf) * (pos ? 1.0f : LAM_NOOBJ);

        if (pos) {
            npos = 1.0f;

            // ---- GIoU vs matched target (raw cx,cy,w,h from tb)
            const float tcx = tb[best_m * 4 + 0], tcy = tb[best_m * 4 + 1];
            const float tw  = tb[best_m * 4 + 2], th  = tb[best_m * 4 + 3];
            const float c1x = tcx - 0.5f * tw, c1y = tcy - 0.5f * th;
            const float c2x = tcx + 0.5f * tw, c2y = tcy + 0.5f * th;
            float iw = fmaxf(fminf(p2x, c2x) - fmaxf(p1x, c1x), 0.0f);
            float ih = fmaxf(fminf(p2y, c2y) - fmaxf(p1y, c1y), 0.0f);
            float inter = iw * ih;
            float uni   = pa + tw * th - inter;
            float iou   = inter / (uni + EPSF);
            float ew = fmaxf(fmaxf(p2x, c2x) - fminf(p1x, c1x), 0.0f);
            float eh = fmaxf(fmaxf(p2y, c2y) - fminf(p1y, c1y), 0.0f);
            float enc = ew * eh;
            float giou = iou - (enc - uni) / (enc + EPSF);
            bbox_sum = 1.0f - giou;

            // ---- focal class loss (class logits read ONLY for pos anchors)
            const float* cl = pred_raw + praw + 5;
            const int    t  = tl[best_m];
            float cs = 0.0f;
            for (int c = 0; c < CNUM; ++c) {
                float x  = cl[c];
                float oh = (c == t) ? 1.0f : 0.0f;
                float ce = bce_logits(x, oh);
                float p  = 1.0f / (1.0f + __expf(-x));
                float pt = (c == t) ? p : (1.0f - p);
                float at = (c == t) ? ALPHA : (1.0f - ALPHA);
                float om = 1.0f - pt;
                cs += at * om * om * ce;
            }
            cls_sum = cs;
        }
    }

    // ---- deterministic block reduction (wave32 shuffle + LDS)
    float v[4] = {obj_sum, bbox_sum, cls_sum, npos};
    #pragma unroll
    for (int k = 0; k < 4; ++k) {
        float x = v[k];
        for (int off = 16; off > 0; off >>= 1)
            x += __shfl_xor(x, off, 32);
        v[k] = x;
    }
    const int wave = tid >> 5, lane = tid & 31;
    if (lane == 0) {
        red[wave * 4 + 0] = v[0]; red[wave * 4 + 1] = v[1];
        red[wave * 4 + 2] = v[2]; red[wave * 4 + 3] = v[3];
    }
    __syncthreads();
    if (tid == 0) {
        float s0 = 0, s1 = 0, s2 = 0, s3 = 0;
        #pragma unroll
        for (int w = 0; w < BLK / 32; ++w) {
            s0 += red[w * 4 + 0]; s1 += red[w * 4 + 1];
            s2 += red[w * 4 + 2]; s3 += red[w * 4 + 3];
        }
        ((float4*)partials)[blockIdx.x] = make_float4(s0, s1, s2, s3);
    }
}

__global__ void __launch_bounds__(BLK)
det_loss_finalize(const float* __restrict__ partials, int nPart,
                  float* __restrict__ out, float invNB)
{
    __shared__ float red[(BLK / 32) * 4];
    float s[4] = {0.0f, 0.0f, 0.0f, 0.0f};
    for (int i = threadIdx.x; i < nPart; i += BLK) {
        float4 p = ((const float4*)partials)[i];
        s[0] += p.x; s[1] += p.y; s[2] += p.z; s[3] += p.w;
    }
    #pragma unroll
    for (int k = 0; k < 4; ++k) {
        float x = s[k];
        for (int off = 16; off > 0; off >>= 1)
            x += __shfl_xor(x, off, 32);
        s[k] = x;
    }
    const int wave = threadIdx.x >> 5, lane = threadIdx.x & 31;
    if (lane == 0) {
        red[wave * 4 + 0] = s[0]; red[wave * 4 + 1] = s[1];
        red[wave * 4 + 2] = s[2]; red[wave * 4 + 3] = s[3];
    }
    __syncthreads();
    if (threadIdx.x == 0) {
        float s0 = 0, s1 = 0, s2 = 0, s3 = 0;
        #pragma unroll
        for (int w = 0; w < BLK / 32; ++w) {
            s0 += red[w * 4 + 0]; s1 += red[w * 4 + 1];
            s2 += red[w * 4 + 2]; s3 += red[w * 4 + 3];
        }
        const float obj_loss   = s0 * invNB;                    // sum/(N*B)
        const float num_pos    = fmaxf(s3, 1.0f);
        const float bbox_loss  = s1 / num_pos * LAM_BBOX;
        const float class_loss = (s2 / (float)CNUM) / num_pos;  // LAM_CLASS = 1
        out[0] = obj_loss + bbox_loss + class_loss;
        out[1] = obj_loss;
        out[2] = bbox_loss;
        out[3] = class_loss;
        out[4] = num_pos;
    }
}

extern "C" void kernel_launch(void* const* d_in, const int* in_sizes, int n_in,
                              void* d_out, int out_size, void* d_ws, size_t ws_size,
                              hipStream_t stream)
{
    const float* pred_raw   = (const float*)d_in[0];  // (B, N, 85) f32
    const float* pred_boxes = (const float*)d_in[1];  // (B, N, 4)  f32
    const float* tgt_boxes  = (const float*)d_in[2];  // (B, M, 4)  f32
    const int*   tgt_labels = (const int*)  d_in[3];  // (B, M)     i32

    const int B = 16;
    const int N = in_sizes[1] / (B * 4);
    const int blocksPerImg = (N + BLK - 1) / BLK;
    const int nPart = B * blocksPerImg;

    float* partials = (float*)d_ws;   // nPart * 4 floats (~25 KB)

    det_loss_main<<<nPart, BLK, 0, stream>>>(pred_raw, pred_boxes, tgt_boxes,
                                             tgt_labels, partials, N, blocksPerImg);
    det_loss_finalize<<<1, BLK, 0, stream>>>(partials, nPart, (float*)d_out,
                                             1.0f / ((float)N * (float)B));
}